// GraphConv_28991029248529
// MI455X (gfx1250) — compile-verified
//
#include <hip/hip_runtime.h>
#include <hip/hip_bf16.h>

typedef __attribute__((ext_vector_type(2))) float v2f;
typedef __attribute__((ext_vector_type(8))) float v8f;

#define F_DIM     128
#define TILE_M    64
#define XS_STRIDE 132                      // pad: stride%64banks==4 -> conflict-free frag reads
#define XS_ELEMS  (TILE_M * XS_STRIDE)     // 8448 floats  (33,792 B)
#define WT_ELEMS  (F_DIM * XS_STRIDE)      // 16896 floats (67,584 B)  -> total LDS 101,376 B

// -----------------------------------------------------------------------------
// Kernel 1: fused dual GEMM.
//   support = X @ W        -> d_ws
//   out     = X @ W_loop+b -> d_out  (scatter accumulates on top)
// Block = 256 threads (8 wave32), 64 rows of X per block (2500 blocks exactly).
// X tile staged via async-to-LDS (ASYNCcnt); weight staged *transposed* so both
// A and B fragments are single conflict-free ds_load_b64 ops feeding
// V_WMMA_F32_16X16X4_F32 (K-loop of 32 per 16x16 tile).
// -----------------------------------------------------------------------------
__global__ __launch_bounds__(256) void gcn_gemm_fused(
    const float* __restrict__ x,
    const float* __restrict__ W,
    const float* __restrict__ Wl,
    const float* __restrict__ bias,
    float* __restrict__ support,
    float* __restrict__ out)
{
    __shared__ float smem[XS_ELEMS + WT_ELEMS];
    float* xs = smem;                // X tile  [64][132]
    float* wt = smem + XS_ELEMS;     // Wt[c][k] transposed weight [128][132]

    const int  tid     = threadIdx.x;
    const long rowBase = (long)blockIdx.x * TILE_M;

    // ---- async-stage X tile: 64x128 f32 = 2048 float4, 8 per thread ----
    {
        const char* gbase = (const char*)(x + rowBase * F_DIM);
#pragma unroll
        for (int j = 0; j < 8; ++j) {
            const int idx = tid + j * 256;                 // float4 index
            const int row = idx >> 5, c4 = idx & 31;
            unsigned lds = (unsigned)(uintptr_t)&xs[row * XS_STRIDE + c4 * 4];
            const char* ga = gbase + (size_t)idx * 16;
            asm volatile("global_load_async_to_lds_b128 %0, %1, off"
                         :: "v"(lds), "v"(ga) : "memory");
        }
    }

    // ---- transpose-stage a 128x128 weight into wt[c][k] ----
    auto stageW = [&](const float* __restrict__ Wg) {
        const float4* wg4 = reinterpret_cast<const float4*>(Wg);
#pragma unroll
        for (int j = 0; j < 16; ++j) {
            const int idx = tid + j * 256;                 // 0..4095 float4
            const int k   = idx >> 5;                      // W row (K)
            const int c4  = idx & 31;                      // float4 column
            float4 v = wg4[idx];
            wt[(c4 * 4 + 0) * XS_STRIDE + k] = v.x;
            wt[(c4 * 4 + 1) * XS_STRIDE + k] = v.y;
            wt[(c4 * 4 + 2) * XS_STRIDE + k] = v.z;
            wt[(c4 * 4 + 3) * XS_STRIDE + k] = v.w;
        }
    };

    stageW(W);
    asm volatile("s_wait_asynccnt 0x0" ::: "memory");   // X tile landed in LDS
    __syncthreads();

    const int wave = tid >> 5;     // 0..7  -> n-tile (16 columns each)
    const int lane = tid & 31;
    const int lhi  = lane >> 4;    // K-pair select (A,B) / M+8 (C)
    const int llo  = lane & 15;    // M (A) / N (B,C)
    const int col  = wave * 16 + llo;
    const float* __restrict__ brow = &wt[col * XS_STRIDE];

    for (int sel = 0; sel < 2; ++sel) {
        if (sel == 1) {
            __syncthreads();       // all waves finished reading W's wt
            stageW(Wl);
            __syncthreads();
        }
        for (int mt = 0; mt < 4; ++mt) {
            v8f acc = {};
            const float* __restrict__ arow = &xs[(mt * 16 + llo) * XS_STRIDE];
#pragma unroll
            for (int k0 = 0; k0 < F_DIM; k0 += 4) {
                const int kk = k0 + 2 * lhi;               // even -> 8B aligned
                v2f a = *reinterpret_cast<const v2f*>(&arow[kk]);  // ds_load_b64
                v2f b = *reinterpret_cast<const v2f*>(&brow[kk]);  // ds_load_b64
                acc = __builtin_amdgcn_wmma_f32_16x16x4_f32(
                        false, a, false, b, (short)0, acc, false, false);
            }
            // VGPR v holds (M = mt*16 + v + 8*lhi, N = col)
            const long r0 = rowBase + mt * 16 + 8 * lhi;
            if (sel == 0) {
#pragma unroll
                for (int v = 0; v < 8; ++v)
                    support[(r0 + v) * F_DIM + col] = acc[v];
            } else {
                const float bv = bias[col];
#pragma unroll
                for (int v = 0; v < 8; ++v)
                    out[(r0 + v) * F_DIM + col] = acc[v] + bv;
            }
        }
    }
}

// -----------------------------------------------------------------------------
// Kernel 2: edge scatter-add.  One wave per (edge, batch) pair:
//   out[b, dst, :] += val * support[b, src, :]
// Pair index forced scalar via readfirstlane -> edge triple comes in on SMEM
// (s_load) instead of 32 redundant vector loads. Lane = 4 features:
// float4 gather + 4x global_atomic_add_f32 (support+out stay L2-resident).
// -----------------------------------------------------------------------------
__global__ __launch_bounds__(256) void gcn_scatter(
    const float* __restrict__ support,
    const float* __restrict__ edge_val,
    const int*   __restrict__ edge_src,
    const int*   __restrict__ edge_dst,
    float* __restrict__ out,
    int nE, int nN)
{
    const long gid  = (long)blockIdx.x * blockDim.x + threadIdx.x;
    const long pair = gid >> 5;                  // wave-uniform
    if (pair >= (long)nE * 4) return;            // wave-uniform branch
    const int lane = threadIdx.x & 31;

    const int p = __builtin_amdgcn_readfirstlane((int)pair);  // force SGPR
    const int e = p >> 2;                        // B = 4
    const int b = p & 3;

    const int   s   = edge_src[e];               // scalar loads (SMEM)
    const int   d   = edge_dst[e];
    const float val = edge_val[e];

    const long srow = ((long)b * nN + s) * F_DIM + lane * 4;
    const long drow = ((long)b * nN + d) * F_DIM + lane * 4;

    float4 m = *reinterpret_cast<const float4*>(&support[srow]);
    float* dp = &out[drow];
    atomicAdd(dp + 0, m.x * val);
    atomicAdd(dp + 1, m.y * val);
    atomicAdd(dp + 2, m.z * val);
    atomicAdd(dp + 3, m.w * val);
}

// -----------------------------------------------------------------------------
extern "C" void kernel_launch(void* const* d_in, const int* in_sizes, int n_in,
                              void* d_out, int out_size, void* d_ws, size_t ws_size,
                              hipStream_t stream) {
    const float* x    = (const float*)d_in[0];
    const float* W    = (const float*)d_in[1];
    const float* Wl   = (const float*)d_in[2];
    const float* bias = (const float*)d_in[3];
    const float* ev   = (const float*)d_in[4];
    const int*   es   = (const int*)d_in[5];
    const int*   ed   = (const int*)d_in[6];
    float* out     = (float*)d_out;
    float* support = (float*)d_ws;

    const int rows = out_size / F_DIM;      // B*N = 160000
    const int nN   = rows / 4;              // N = 40000 (B = 4)
    const int nE   = in_sizes[4];           // E = 640000

    // 1) dual GEMM: support -> ws, out = X@Wl + bias
    gcn_gemm_fused<<<rows / TILE_M, 256, 0, stream>>>(x, W, Wl, bias, support, out);

    // 2) scatter-add over edges (one wave per (e,b))
    const long threads = (long)nE * 4 * 32;
    const int  blocks  = (int)((threads + 255) / 256);
    gcn_scatter<<<blocks, 256, 0, stream>>>(support, ev, es, ed, out, nE, nN);
}